// ICP_28183575396451
// MI455X (gfx1250) — compile-verified
//
#include <hip/hip_runtime.h>
#include <math.h>

typedef __attribute__((ext_vector_type(2))) float v2f;
typedef __attribute__((ext_vector_type(8))) float v8f;

#define ICP_STEPS 8

// ---------------------------------------------------------------------------
// init: copy source -> P (current point set), precompute |t_m|^2
// ---------------------------------------------------------------------------
__global__ void icp_init_kernel(const float* __restrict__ src,
                                const float* __restrict__ tgt,
                                float* __restrict__ P,
                                float* __restrict__ tnorm,
                                int BN, int BM) {
    int i = blockIdx.x * blockDim.x + threadIdx.x;
    if (i < BN) {
        P[3 * i + 0] = src[3 * i + 0];
        P[3 * i + 1] = src[3 * i + 1];
        P[3 * i + 2] = src[3 * i + 2];
    }
    if (i < BM) {
        float x = tgt[3 * i + 0], y = tgt[3 * i + 1], z = tgt[3 * i + 2];
        tnorm[i] = x * x + y * y + z * z;
    }
}

// ---------------------------------------------------------------------------
// kNN via V_WMMA_F32_16X16X4_F32: one wave handles a 16-row source tile,
// streams all target tiles of 16 columns. score = |t|^2 - 2*(s.t).
// ---------------------------------------------------------------------------
__global__ void __launch_bounds__(128)
icp_knn_kernel(const float* __restrict__ P, const float* __restrict__ T,
               const float* __restrict__ tnorm, int* __restrict__ idx,
               int N, int M, int nwaves) {
    int gwave = (int)((blockIdx.x * blockDim.x + threadIdx.x) >> 5);
    int lane  = (int)(threadIdx.x & 31);
    if (gwave >= nwaves) return;              // uniform per wave; EXEC stays all-ones

    int tilesPerBatch = N >> 4;
    int b    = gwave / tilesPerBatch;
    int tile = gwave - b * tilesPerBatch;
    int row0 = tile << 4;

    const float* Pb = P + (size_t)b * N * 3;
    const float* Tb = T + (size_t)b * M * 3;
    const float* tn = tnorm + (size_t)b * M;

    int  r  = lane & 15;
    bool hi = lane >= 16;

    // A matrix (16x4): row = lane&15; lanes 0-15 carry K=0,1 ({x,y}),
    // lanes 16-31 carry K=2,3 ({z,0}).
    const float* ps = Pb + (size_t)(row0 + r) * 3;
    float px = ps[0], py = ps[1], pz = ps[2];
    v2f a;
    a.x = hi ? pz : px;
    a.y = hi ? 0.0f : py;

    float bestv[8];
    int   besti[8];
#pragma unroll
    for (int i = 0; i < 8; ++i) { bestv[i] = 3.4e38f; besti[i] = 0; }

    for (int t0 = 0; t0 < M; t0 += 16) {
        // B matrix (4x16): col = lane&15, same half-wave K split as A.
        const float* ts = Tb + (size_t)(t0 + r) * 3;
        float qx = ts[0], qy = ts[1], qz = ts[2];
        v2f bm;
        bm.x = hi ? qz : qx;
        bm.y = hi ? 0.0f : qy;
        float tnv = tn[t0 + r];

        v8f c = {0.f, 0.f, 0.f, 0.f, 0.f, 0.f, 0.f, 0.f};
        c = __builtin_amdgcn_wmma_f32_16x16x4_f32(
                /*neg_a=*/false, a, /*neg_b=*/false, bm,
                /*c_mod=*/(short)0, c, /*reuse_a=*/false, /*reuse_b=*/false);

        int cidx = t0 + r;
#pragma unroll
        for (int i = 0; i < 8; ++i) {
            float s = tnv - 2.0f * c[i];
            if (s < bestv[i]) { bestv[i] = s; besti[i] = cidx; }
        }
    }

    // min-reduce over the 16 columns held across each half-wave
#pragma unroll
    for (int m = 8; m >= 1; m >>= 1) {
#pragma unroll
        for (int i = 0; i < 8; ++i) {
            float ov = __shfl_xor(bestv[i], m, 32);
            int   oi = __shfl_xor(besti[i], m, 32);
            if (ov < bestv[i] || (ov == bestv[i] && oi < besti[i])) {
                bestv[i] = ov; besti[i] = oi;
            }
        }
    }

    if (r == 0) {
        int  base = row0 + (hi ? 8 : 0);     // VGPR i: rows i (lo half) / i+8 (hi half)
        int* ib   = idx + (size_t)b * N;
#pragma unroll
        for (int i = 0; i < 8; ++i) ib[base + i] = besti[i];
    }
}

// ---------------------------------------------------------------------------
// deterministic partial sums: Sp(3), Sq(3), Sum p q^T (9) per block per batch
// q = Q[idx[n]] if idx != nullptr, else q = Q[n]
// ---------------------------------------------------------------------------
__global__ void __launch_bounds__(256)
icp_sums_kernel(const float* __restrict__ P, const float* __restrict__ Q,
                const int* __restrict__ idx, float* __restrict__ partial,
                int N, int M) {
    int b = blockIdx.y;
    int n = blockIdx.x * blockDim.x + threadIdx.x;

    float v[15];
#pragma unroll
    for (int k = 0; k < 15; ++k) v[k] = 0.f;

    if (n < N) {
        const float* p = P + ((size_t)b * N + n) * 3;
        const float* q;
        if (idx) q = Q + ((size_t)b * M + idx[(size_t)b * N + n]) * 3;
        else     q = Q + ((size_t)b * N + n) * 3;
        float p0 = p[0], p1 = p[1], p2 = p[2];
        float q0 = q[0], q1 = q[1], q2 = q[2];
        v[0] = p0; v[1] = p1; v[2] = p2;
        v[3] = q0; v[4] = q1; v[5] = q2;
        v[6] = p0 * q0; v[7]  = p0 * q1; v[8]  = p0 * q2;
        v[9] = p1 * q0; v[10] = p1 * q1; v[11] = p1 * q2;
        v[12] = p2 * q0; v[13] = p2 * q1; v[14] = p2 * q2;
    }

#pragma unroll
    for (int m = 16; m >= 1; m >>= 1) {
#pragma unroll
        for (int k = 0; k < 15; ++k) v[k] += __shfl_xor(v[k], m, 32);
    }

    __shared__ float smem[8 * 15];
    int wave = (int)(threadIdx.x >> 5);
    int lane = (int)(threadIdx.x & 31);
    if (lane == 0) {
#pragma unroll
        for (int k = 0; k < 15; ++k) smem[wave * 15 + k] = v[k];
    }
    __syncthreads();
    if (threadIdx.x < 15) {
        float s = 0.f;
        int nw = (int)(blockDim.x >> 5);
        for (int w = 0; w < nw; ++w) s += smem[w * 15 + (int)threadIdx.x];
        partial[((size_t)b * gridDim.x + blockIdx.x) * 16 + threadIdx.x] = s;
    }
}

// ---------------------------------------------------------------------------
// Kabsch via 3x3 Jacobi-SVD; one thread per batch.
// ---------------------------------------------------------------------------
__device__ __forceinline__ float icp_det3(const float Mx[3][3]) {
    return Mx[0][0] * (Mx[1][1] * Mx[2][2] - Mx[1][2] * Mx[2][1])
         - Mx[0][1] * (Mx[1][0] * Mx[2][2] - Mx[1][2] * Mx[2][0])
         + Mx[0][2] * (Mx[1][0] * Mx[2][1] - Mx[1][1] * Mx[2][0]);
}

__global__ void icp_kabsch_kernel(const float* __restrict__ partial, int nblk,
                                  float* __restrict__ RT, float* __restrict__ out,
                                  int N, int B) {
    int b = (int)threadIdx.x;
    if (b >= B) return;

    float s[15];
#pragma unroll
    for (int k = 0; k < 15; ++k) s[k] = 0.f;
    for (int blk = 0; blk < nblk; ++blk)
        for (int k = 0; k < 15; ++k)
            s[k] += partial[((size_t)b * nblk + blk) * 16 + k];

    float invN = 1.0f / (float)N;
    float pm[3] = { s[0] * invN, s[1] * invN, s[2] * invN };
    float qm[3] = { s[3] * invN, s[4] * invN, s[5] * invN };

    float H[3][3];
    for (int i = 0; i < 3; ++i)
        for (int j = 0; j < 3; ++j)
            H[i][j] = s[6 + 3 * i + j] - (float)N * pm[i] * qm[j];

    // A = H^T H (symmetric)
    float A[3][3];
    for (int i = 0; i < 3; ++i)
        for (int j = 0; j < 3; ++j)
            A[i][j] = H[0][i] * H[0][j] + H[1][i] * H[1][j] + H[2][i] * H[2][j];

    float V[3][3] = { {1, 0, 0}, {0, 1, 0}, {0, 0, 1} };
    const int PP[3] = { 0, 0, 1 }, QQ[3] = { 1, 2, 2 };
    for (int sweep = 0; sweep < 15; ++sweep) {
        for (int k = 0; k < 3; ++k) {
            int p = PP[k], q = QQ[k];
            float apq = A[p][q];
            if (fabsf(apq) < 1e-30f) continue;
            float theta = (A[q][q] - A[p][p]) / (2.0f * apq);
            float t = copysignf(1.0f, theta) / (fabsf(theta) + sqrtf(theta * theta + 1.0f));
            float c = 1.0f / sqrtf(t * t + 1.0f);
            float ss = t * c;
            float app = A[p][p], aqq = A[q][q];
            A[p][p] = app - t * apq;
            A[q][q] = aqq + t * apq;
            A[p][q] = 0.f; A[q][p] = 0.f;
            int r2 = 3 - p - q;
            float arp = A[r2][p], arq = A[r2][q];
            A[r2][p] = c * arp - ss * arq; A[p][r2] = A[r2][p];
            A[r2][q] = ss * arp + c * arq; A[q][r2] = A[r2][q];
            for (int rr = 0; rr < 3; ++rr) {
                float vrp = V[rr][p], vrq = V[rr][q];
                V[rr][p] = c * vrp - ss * vrq;
                V[rr][q] = ss * vrp + c * vrq;
            }
        }
    }

    float w[3] = { A[0][0], A[1][1], A[2][2] };
    for (int i = 0; i < 2; ++i)
        for (int j = i + 1; j < 3; ++j)
            if (w[j] > w[i]) {
                float tw = w[i]; w[i] = w[j]; w[j] = tw;
                for (int rr = 0; rr < 3; ++rr) {
                    float tv = V[rr][i]; V[rr][i] = V[rr][j]; V[rr][j] = tv;
                }
            }

    float sg[3];
    for (int i = 0; i < 3; ++i) sg[i] = sqrtf(fmaxf(w[i], 0.f));

    float U[3][3];
    for (int i = 0; i < 3; ++i) {
        float hx = H[0][0] * V[0][i] + H[0][1] * V[1][i] + H[0][2] * V[2][i];
        float hy = H[1][0] * V[0][i] + H[1][1] * V[1][i] + H[1][2] * V[2][i];
        float hz = H[2][0] * V[0][i] + H[2][1] * V[1][i] + H[2][2] * V[2][i];
        bool ok = sg[i] > 1e-12f * fmaxf(sg[0], 1e-30f);
        if (ok) {
            float is = 1.0f / sg[i];
            U[0][i] = hx * is; U[1][i] = hy * is; U[2][i] = hz * is;
        } else if (i == 2) {
            U[0][2] = U[1][0] * U[2][1] - U[2][0] * U[1][1];
            U[1][2] = U[2][0] * U[0][1] - U[0][0] * U[2][1];
            U[2][2] = U[0][0] * U[1][1] - U[1][0] * U[0][1];
        } else {
            U[0][i] = (i == 0) ? 1.f : 0.f;
            U[1][i] = (i == 1) ? 1.f : 0.f;
            U[2][i] = 0.f;
        }
    }

    float d = icp_det3(V) * icp_det3(U);
    float ds[3] = { 1.f, 1.f, d };

    float R[3][3];
    for (int i = 0; i < 3; ++i)
        for (int j = 0; j < 3; ++j)
            R[i][j] = V[i][0] * ds[0] * U[j][0] + V[i][1] * ds[1] * U[j][1] + V[i][2] * ds[2] * U[j][2];

    float t[3];
    for (int i = 0; i < 3; ++i)
        t[i] = qm[i] - (R[i][0] * pm[0] + R[i][1] * pm[1] + R[i][2] * pm[2]);

    float* rt = RT + (size_t)b * 12;
    rt[0] = t[0]; rt[1] = t[1]; rt[2] = t[2];
    for (int i = 0; i < 3; ++i)
        for (int j = 0; j < 3; ++j)
            rt[3 + 3 * i + j] = R[i][j];

    if (out) {
        float* o = out + (size_t)b * 12;
        o[0] = t[0]; o[1] = t[1]; o[2] = t[2];
        for (int i = 0; i < 3; ++i)
            for (int j = 0; j < 3; ++j)
                o[3 + 3 * i + j] = R[i][j];
    }
}

// ---------------------------------------------------------------------------
// apply rigid transform in place: P <- R P + t
// ---------------------------------------------------------------------------
__global__ void icp_transform_kernel(float* __restrict__ P, const float* __restrict__ RT,
                                     int N, int B) {
    int i = blockIdx.x * blockDim.x + threadIdx.x;
    if (i >= B * N) return;
    int b = i / N;
    const float* rt = RT + (size_t)b * 12;
    float* p = P + (size_t)i * 3;
    float x = p[0], y = p[1], z = p[2];
    float nx = rt[3] * x + rt[4]  * y + rt[5]  * z + rt[0];
    float ny = rt[6] * x + rt[7]  * y + rt[8]  * z + rt[1];
    float nz = rt[9] * x + rt[10] * y + rt[11] * z + rt[2];
    p[0] = nx; p[1] = ny; p[2] = nz;
}

// ---------------------------------------------------------------------------
extern "C" void kernel_launch(void* const* d_in, const int* in_sizes, int n_in,
                              void* d_out, int out_size, void* d_ws, size_t ws_size,
                              hipStream_t stream) {
    (void)n_in; (void)out_size; (void)ws_size;
    const float* src = (const float*)d_in[0];
    const float* tgt = (const float*)d_in[1];

    const int N = 4096;
    const int B = in_sizes[0] / (N * 3);
    const int M = in_sizes[1] / (B * 3);

    // workspace carve-up (256B aligned)
    char*  ws  = (char*)d_ws;
    size_t off = 0;
    float* P      = (float*)(ws + off); off += ((size_t)B * N * 3 * 4 + 255) & ~255ULL;
    float* tnorm  = (float*)(ws + off); off += ((size_t)B * M * 4 + 255) & ~255ULL;
    int*   idx    = (int*)  (ws + off); off += ((size_t)B * N * 4 + 255) & ~255ULL;
    const int nblkSums = (N + 255) / 256;
    float* partial = (float*)(ws + off); off += ((size_t)B * nblkSums * 16 * 4 + 255) & ~255ULL;
    float* RT     = (float*)(ws + off); off += ((size_t)B * 12 * 4 + 255) & ~255ULL;

    // init: P = source, tnorm = |target|^2
    {
        int total = (B * N > B * M) ? B * N : B * M;
        icp_init_kernel<<<(total + 255) / 256, 256, 0, stream>>>(src, tgt, P, tnorm, B * N, B * M);
    }

    const int nwaves   = B * (N >> 4);
    const int knnBlks  = (nwaves * 32 + 127) / 128;
    dim3 sumsGrid(nblkSums, B);

    for (int step = 0; step < ICP_STEPS; ++step) {
        icp_knn_kernel<<<knnBlks, 128, 0, stream>>>(P, tgt, tnorm, idx, N, M, nwaves);
        icp_sums_kernel<<<sumsGrid, 256, 0, stream>>>(P, tgt, idx, partial, N, M);
        icp_kabsch_kernel<<<1, 32, 0, stream>>>(partial, nblkSums, RT, nullptr, N, B);
        icp_transform_kernel<<<(B * N + 255) / 256, 256, 0, stream>>>(P, RT, N, B);
    }

    // final: Kabsch from original source to converged cloud, write (B,12) output
    icp_sums_kernel<<<sumsGrid, 256, 0, stream>>>(src, P, nullptr, partial, N, M);
    icp_kabsch_kernel<<<1, 32, 0, stream>>>(partial, nblkSums, RT, (float*)d_out, N, B);
}